// DirSAGE_57432302682549
// MI455X (gfx1250) — compile-verified
//
#include <hip/hip_runtime.h>

typedef __attribute__((ext_vector_type(2))) float v2f;
typedef __attribute__((ext_vector_type(8))) float v8f;

#define NN 50000
#define EE 800000
#define DD 128
#define OUTD 64

// ---------------- utility kernels ----------------

__global__ void zero_f4_kernel(float4* __restrict__ p, int n4) {
    int t = blockIdx.x * blockDim.x + threadIdx.x;
    if (t < n4) p[t] = make_float4(0.f, 0.f, 0.f, 0.f);
}

__device__ __forceinline__ void atomAddF(float* p, float v) {
    unsafeAtomicAdd(p, v);   // lowers to global_atomic_add_f32 on gfx1250
}

__global__ void degree_kernel(const int* __restrict__ src, const int* __restrict__ dst,
                              float* __restrict__ degS, float* __restrict__ degD) {
    int e = blockIdx.x * blockDim.x + threadIdx.x;
    if (e < EE) {
        atomAddF(&degD[dst[e]], 1.0f);
        atomAddF(&degS[src[e]], 1.0f);
    }
}

// sDst = (1-a)/max(degD,1)  (scale for src->dst mean agg)
// sSrc = a/max(degS,1)      (scale for dst->src mean agg)
__global__ void scale_kernel(const float* __restrict__ degS, const float* __restrict__ degD,
                             const float* __restrict__ alpha,
                             float* __restrict__ sSrc, float* __restrict__ sDst) {
    int n = blockIdx.x * blockDim.x + threadIdx.x;
    if (n < NN) {
        float a = alpha[0];
        float dd = degD[n]; dd = dd < 1.f ? 1.f : dd;
        float ds = degS[n]; ds = ds < 1.f ? 1.f : ds;
        sDst[n] = (1.0f - a) / dd;
        sSrc[n] = a / ds;
    }
}

// one wave per edge: lanes cover the 128-float feature row (float4 chunks)
__global__ void scatter_kernel(const float* __restrict__ H,
                               const int* __restrict__ src, const int* __restrict__ dst,
                               float* __restrict__ aggStd, float* __restrict__ aggDts) {
    int t = blockIdx.x * blockDim.x + threadIdx.x;
    if (t >= EE * 32) return;
    int e = t >> 5;
    int c = (t & 31) << 2;
    int s = src[e], d = dst[e];
    const float4 hs = *(const float4*)(H + s * DD + c);
    const float4 hd = *(const float4*)(H + d * DD + c);
    float* ps = aggStd + d * DD + c;
    atomAddF(ps + 0, hs.x); atomAddF(ps + 1, hs.y);
    atomAddF(ps + 2, hs.z); atomAddF(ps + 3, hs.w);
    float* pd = aggDts + s * DD + c;
    atomAddF(pd + 0, hd.x); atomAddF(pd + 1, hd.y);
    atomAddF(pd + 2, hd.z); atomAddF(pd + 3, hd.w);
}

__global__ void normalize_kernel(float* __restrict__ aggStd, float* __restrict__ aggDts,
                                 const float* __restrict__ sDst, const float* __restrict__ sSrc) {
    int t = blockIdx.x * blockDim.x + threadIdx.x;
    if (t >= NN * 32) return;
    int n = t >> 5;
    int c = (t & 31) << 2;
    float s1 = sDst[n];
    float4* p = (float4*)(aggStd + n * DD + c);
    float4 v = *p; v.x *= s1; v.y *= s1; v.z *= s1; v.w *= s1; *p = v;
    float s2 = sSrc[n];
    float4* q = (float4*)(aggDts + n * DD + c);
    float4 w = *q; w.x *= s2; w.y *= s2; w.z *= s2; w.w *= s2; *q = w;
}

// ---------------- WMMA GEMM ----------------
// One wave computes a 16x16 f32 tile with V_WMMA_F32_16X16X4_F32, K=128 per phase.
// A (16x4/step): lane<16 holds K=2k,2k+1 of row (m0+lane); lane>=16 holds K=2k+2,2k+3.
// B (4x16/step): vgpr0 = row K (lanes 0-15) / K+2 (lanes 16-31); vgpr1 = K+1 / K+3.
__device__ __forceinline__ v8f wmma_phase128(const float* __restrict__ A,
                                             const float* __restrict__ W, int ldb,
                                             int m0, int n0, int l, int half, v8f c) {
    const float* arow = A + (m0 + l) * DD + 2 * half;
    const float* bcol = W + n0 + l + (2 * half) * ldb;
#pragma unroll
    for (int kk = 0; kk < 128; kk += 4) {
        v2f a = *(const v2f*)(arow + kk);
        v2f b;
        b.x = bcol[kk * ldb];
        b.y = bcol[(kk + 1) * ldb];
        c = __builtin_amdgcn_wmma_f32_16x16x4_f32(false, a, false, b, (short)0, c,
                                                  false, false);
    }
    return c;
}

// H' = relu(h@W0 + agg_std_scaled@W1 + agg_dts_scaled@W2 + bias); Jk = max(Jk, H')
__global__ void __launch_bounds__(256) layer_gemm_kernel(
    const float* __restrict__ A0, const float* __restrict__ A1, const float* __restrict__ A2,
    const float* __restrict__ W0, const float* __restrict__ W1, const float* __restrict__ W2,
    const float* __restrict__ b0, const float* __restrict__ b1, const float* __restrict__ b2,
    const float* __restrict__ alpha,
    float* __restrict__ Hout, float* __restrict__ Jk, int first)
{
    const int wave = threadIdx.x >> 5;
    const int lane = threadIdx.x & 31;
    const int half = lane >> 4;
    const int l = lane & 15;
    const int m0 = blockIdx.x * 16;
    const int n0 = wave * 16;

    v8f c = {0.f, 0.f, 0.f, 0.f, 0.f, 0.f, 0.f, 0.f};
    c = wmma_phase128(A0, W0, DD, m0, n0, l, half, c);   // self term
    c = wmma_phase128(A1, W1, DD, m0, n0, l, half, c);   // (1-a)*agg_std (prescaled)
    c = wmma_phase128(A2, W2, DD, m0, n0, l, half, c);   // a*agg_dts    (prescaled)

    float a = alpha[0];
    int col = n0 + l;
    float bias = b0[col] + (1.0f - a) * b1[col] + a * b2[col];

#pragma unroll
    for (int r = 0; r < 8; ++r) {
        int row = m0 + r + 8 * half;
        float v = c[r] + bias;
        v = v > 0.f ? v : 0.f;
        Hout[row * DD + col] = v;
        float j = first ? v : fmaxf(Jk[row * DD + col], v);
        Jk[row * DD + col] = j;
    }
}

// out = Jk @ Wlin + blin   (N x 128 @ 128 x 64)
__global__ void __launch_bounds__(128) final_gemm_kernel(
    const float* __restrict__ Jk, const float* __restrict__ Wlin,
    const float* __restrict__ blin, float* __restrict__ Out)
{
    const int wave = threadIdx.x >> 5;
    const int lane = threadIdx.x & 31;
    const int half = lane >> 4;
    const int l = lane & 15;
    const int m0 = blockIdx.x * 16;
    const int n0 = wave * 16;

    v8f c = {0.f, 0.f, 0.f, 0.f, 0.f, 0.f, 0.f, 0.f};
    c = wmma_phase128(Jk, Wlin, OUTD, m0, n0, l, half, c);

    int col = n0 + l;
    float bias = blin[col];
#pragma unroll
    for (int r = 0; r < 8; ++r) {
        int row = m0 + r + 8 * half;
        Out[row * OUTD + col] = c[r] + bias;
    }
}

// ---------------- host launcher ----------------

extern "C" void kernel_launch(void* const* d_in, const int* in_sizes, int n_in,
                              void* d_out, int out_size, void* d_ws, size_t ws_size,
                              hipStream_t stream) {
    const float* x     = (const float*)d_in[0];
    const int*   ei    = (const int*)d_in[1];
    const int*   src   = ei;          // edge_index[0]
    const int*   dst   = ei + EE;     // edge_index[1]
    const float* Wself = (const float*)d_in[2];
    const float* bself = (const float*)d_in[3];
    const float* Wstd  = (const float*)d_in[4];
    const float* bstd  = (const float*)d_in[5];
    const float* Wdts  = (const float*)d_in[6];
    const float* bdts  = (const float*)d_in[7];
    const float* Wlin  = (const float*)d_in[8];
    const float* blin  = (const float*)d_in[9];
    const float* alpha = (const float*)d_in[10];
    float* out = (float*)d_out;

    char* ws = (char*)d_ws;
    size_t off = 0;
    auto alloc = [&](size_t bytes) -> float* {
        float* p = (float*)(ws + off);
        off += (bytes + 255) & ~(size_t)255;
        return p;
    };
    float* hA     = alloc((size_t)NN * DD * 4);
    float* hB     = alloc((size_t)NN * DD * 4);
    float* jk     = alloc((size_t)NN * DD * 4);
    float* aggStd = alloc((size_t)NN * DD * 4);   // contiguous with aggDts
    float* aggDts = alloc((size_t)NN * DD * 4);
    float* deg    = alloc((size_t)2 * NN * 4);    // [degD | degS]
    float* scl    = alloc((size_t)2 * NN * 4);    // [sDst | sSrc]
    float* degD = deg,  *degS = deg + NN;
    float* sDst = scl,  *sSrc = scl + NN;

    const int TB = 256;

    // degrees + per-node scale factors (constant across layers)
    zero_f4_kernel<<<(2 * NN / 4 + TB - 1) / TB, TB, 0, stream>>>((float4*)deg, 2 * NN / 4);
    degree_kernel<<<(EE + TB - 1) / TB, TB, 0, stream>>>(src, dst, degS, degD);
    scale_kernel<<<(NN + TB - 1) / TB, TB, 0, stream>>>(degS, degD, alpha, sSrc, sDst);

    const float* Hcur = x;
    float* bufs[2] = {hA, hB};
    for (int i = 0; i < 3; ++i) {
        zero_f4_kernel<<<(2 * NN * DD / 4 + TB - 1) / TB, TB, 0, stream>>>(
            (float4*)aggStd, 2 * NN * DD / 4);
        scatter_kernel<<<(EE * 32 + TB - 1) / TB, TB, 0, stream>>>(
            Hcur, src, dst, aggStd, aggDts);
        normalize_kernel<<<(NN * 32 + TB - 1) / TB, TB, 0, stream>>>(
            aggStd, aggDts, sDst, sSrc);
        float* Hnext = bufs[i & 1];
        layer_gemm_kernel<<<NN / 16, 256, 0, stream>>>(
            Hcur, aggStd, aggDts,
            Wself + (size_t)i * DD * DD, Wstd + (size_t)i * DD * DD, Wdts + (size_t)i * DD * DD,
            bself + (size_t)i * DD, bstd + (size_t)i * DD, bdts + (size_t)i * DD,
            alpha, Hnext, jk, i == 0 ? 1 : 0);
        Hcur = Hnext;
    }

    final_gemm_kernel<<<NN / 16, 128, 0, stream>>>(jk, Wlin, blin, out);
}